// NDDE_1D_59390807769589
// MI455X (gfx1250) — compile-verified
//
#include <hip/hip_runtime.h>
#include <hip/hip_bf16.h>
#include <math.h>

typedef float v2f __attribute__((ext_vector_type(2)));
typedef float v8f __attribute__((ext_vector_type(8)));
typedef int   v2i __attribute__((ext_vector_type(2)));

#define D     512
#define RING  128          // > 101+1 delay span; power of two for cheap mod
#define NTAU  100
#define NWG   32           // 512 rows / 16 rows per WG
#define TPB   256          // 8 wave32s per WG

// gfx1250 async global->LDS copy (ASYNCcnt-tracked). Signature learned from
// the round-2 diagnostic: param0 = AS(1) v2i* (non-const), then LDS ptr,
// imm offset, cpol. Guarded so a different-toolchain spelling still compiles
// via the ds_store fallback.
#if defined(__AMDGCN__) && __has_builtin(__builtin_amdgcn_global_load_async_to_lds_b64)
#define HAVE_ASYNC_LDS 1
typedef __attribute__((address_space(1))) v2i g_v2i;   // global int2
typedef __attribute__((address_space(3))) v2i l_v2i;   // LDS int2
__device__ __forceinline__ void wait_async0() {
#if __has_builtin(__builtin_amdgcn_s_wait_asynccnt)
  __builtin_amdgcn_s_wait_asynccnt(0);
#else
  asm volatile("s_wait_asynccnt 0x0" ::: "memory");
#endif
}
#else
#define HAVE_ASYNC_LDS 0
#endif

// ---------------------------------------------------------------------------
// Init kernel: stream-ordered before the solver every launch (capture safe).
// Seeds trajectory column 0 and ring slot 0 with x0, zeroes the barrier
// counter so every graph replay is deterministic.
// ---------------------------------------------------------------------------
__global__ void ndde_init(const float* __restrict__ x0, float* __restrict__ out,
                          int ncols, float* __restrict__ ring,
                          unsigned* __restrict__ cnt) {
  int i = blockIdx.x * blockDim.x + threadIdx.x;
  if (i < D) {
    float v = x0[i];
    ring[i] = v;                      // ring slot 0 = x(t<=0)
    out[(size_t)i * ncols] = v;       // trajectory column 0
  }
  if (i == 0) *cnt = 0u;
}

// ---------------------------------------------------------------------------
// Persistent cooperative solver. 32 WGs; WG g owns output rows [16g,16g+16).
// Wave w: matrix = w&1 (W1/W2), K-slice = 128*(w>>1). Each wave keeps its
// 16x128 f32 weight slice resident in VGPRs (32 v2f A-fragments) for all N
// steps and accumulates with v_wmma_f32_16x16x4_f32. Per step, the current
// and delayed state vectors are staged into LDS once (async copy) and the
// WMMA B-fragments are fed from LDS.
// ---------------------------------------------------------------------------
__global__ void __launch_bounds__(TPB) ndde_solve(
    const float* __restrict__ tau,
    const float* __restrict__ W1, const float* __restrict__ W2,
    const float* __restrict__ b,  const int* __restrict__ Np,
    float* __restrict__ out, float* __restrict__ ring,
    unsigned* __restrict__ cnt)
{
  const int   N     = *Np;
  const int   ncols = N + 1;
  const float dt    = 0.01f * tau[0];

  const int tid  = threadIdx.x;
  const int lane = tid & 31;
  const int wave = tid >> 5;
  const int mat  = wave & 1;            // 0 -> W1 (current x), 1 -> W2 (delayed y)
  const int K0   = (wave >> 1) * 128;   // this wave's K-slice
  const int ROW0 = blockIdx.x * 16;     // this WG's row stripe

  // A-fragment lane mapping for 16x16x4 f32 WMMA:
  //   lanes 0-15 : M = lane,    VGPR0/1 = K+0 / K+1
  //   lanes 16-31: M = lane-16, VGPR0/1 = K+2 / K+3
  const int rowl = ROW0 + (lane & 15);
  const int koff = (lane >> 4) * 2;

  const float* __restrict__ Wm = mat ? W2 : W1;

  // ---- one-time weight preload: 16x128 slice -> 32 A-fragments (64 VGPRs) --
  v2f a[32];
#pragma unroll
  for (int f = 0; f < 32; ++f) {
    int k = K0 + 4 * f + koff;
    a[f] = *(const v2f*)(Wm + (size_t)rowl * D + k);
  }

  const float bias = (tid < 16) ? b[ROW0 + tid] : 0.0f;   // hoisted

  __shared__ float acc[16];             // W1@x + W2@y + b for this row stripe
  __shared__ float xls[D];              // staged x_j
  __shared__ float yls[D];              // staged delayed state

  for (int j = 0; j < N; ++j) {
    const int xs = j & (RING - 1);                          // x_j slot
    const int ys = (j >= NTAU ? j - NTAU : 0) & (RING - 1); // delayed slot
    const int ws = (j + 1) & (RING - 1);                    // x_{j+1} slot

    // ---- stage x and y (2 KB each) into LDS: one b64 per thread per vector
    {
      const float* xsrc = ring + (size_t)xs * D;
      const float* ysrc = ring + (size_t)ys * D;
      const int i = tid * 2;
#if HAVE_ASYNC_LDS
      __builtin_amdgcn_global_load_async_to_lds_b64(
          (g_v2i*)(xsrc + i), (l_v2i*)&xls[i], 0, 0);
      __builtin_amdgcn_global_load_async_to_lds_b64(
          (g_v2i*)(ysrc + i), (l_v2i*)&yls[i], 0, 0);
#else
      *(v2f*)&xls[i] = *(const v2f*)(xsrc + i);
      *(v2f*)&yls[i] = *(const v2f*)(ysrc + i);
#endif
    }
    if (tid < 16) acc[tid] = bias;
#if HAVE_ASYNC_LDS
    wait_async0();                       // this wave's async LDS writes done
#endif
    __syncthreads();                     // publish staged vectors + acc init

    // ---- 32 chained f32 WMMAs fed from LDS --------------------------------
    const float* vec = (mat ? yls : xls);
    v8f c = {};
#pragma unroll
    for (int f = 0; f < 32; ++f) {
      v2f bb = *(const v2f*)(vec + K0 + 4 * f + koff);   // ds_load_b64
      c = __builtin_amdgcn_wmma_f32_16x16x4_f32(false, a[f], false, bb,
                                                (short)0, c, false, false);
    }

    // Columns of C are identical; lane 0 carries rows 0-7, lane 16 rows 8-15.
    // Reduce the 8 K-slice partials across waves with LDS float atomics.
    if ((lane & 15) == 0) {
      int rb = (lane >> 4) * 8;
#pragma unroll
      for (int v = 0; v < 8; ++v) atomicAdd(&acc[rb + v], c[v]);
    }
    __syncthreads();

    // ---- Euler update for this WG's 16 rows -------------------------------
    if (tid < 16) {
      int   row = ROW0 + tid;
      float xn  = xls[row] + dt * tanhf(acc[tid]);
      ring[(size_t)ws * D + row]         = xn;
      out[(size_t)row * ncols + (j + 1)] = xn;
    }
    __syncthreads();

    // ---- grid-wide step barrier: monotonic counter + release/acquire ------
    if (tid == 0) {
      __threadfence();                          // release ring writes to L2
      atomicAdd(cnt, 1u);
      const unsigned target = (unsigned)NWG * (unsigned)(j + 1);
      while (__hip_atomic_load(cnt, __ATOMIC_RELAXED,
                               __HIP_MEMORY_SCOPE_AGENT) < target)
        __builtin_amdgcn_s_sleep(1);
      __threadfence();                          // acquire other WGs' writes
    }
    __syncthreads();
  }
}

// ---------------------------------------------------------------------------
extern "C" void kernel_launch(void* const* d_in, const int* in_sizes, int n_in,
                              void* d_out, int out_size, void* d_ws, size_t ws_size,
                              hipStream_t stream) {
  const float* x0  = (const float*)d_in[0];
  const float* tau = (const float*)d_in[1];
  const float* W1  = (const float*)d_in[2];
  const float* W2  = (const float*)d_in[3];
  const float* b   = (const float*)d_in[4];
  const int*   Np  = (const int*)d_in[5];
  float* out = (float*)d_out;

  unsigned* cnt  = (unsigned*)d_ws;                 // barrier counter
  float*    ring = (float*)((char*)d_ws + 256);     // 128*512 f32 = 256 KB

  const int ncols = out_size / D;                   // N + 1

  ndde_init<<<2, 256, 0, stream>>>(x0, out, ncols, ring, cnt);
  ndde_solve<<<NWG, TPB, 0, stream>>>(tau, W1, W2, b, Np, out, ring, cnt);
}